// Upper_encoder_32349693674040
// MI455X (gfx1250) — compile-verified
//
#include <hip/hip_runtime.h>
#include <hip/hip_bf16.h>

// ---------------------------------------------------------------------------
// 2-layer GCN (PyG GCNConv + LayerNorm + ReLU) for MI455X / gfx1250.
//   N = 100000 nodes, C = 256 channels, E = 1.6M edges (+N self loops)
// Phases per layer:
//   1) xw = z @ W      -- WMMA f32 16x16x4, A stripe staged in LDS via TDM
//   2) agg = scatter-add over edges of xw[row] * isd[row]*isd[col]
//   3) z   = relu(layernorm(agg + b))   (in place on the output buffer)
// ---------------------------------------------------------------------------

#define C_DIM 256
#define LDS_STRIDE 260   // 256 + 4-dword TDM pad -> conflict-free A reads

typedef __attribute__((ext_vector_type(2))) float v2f;
typedef __attribute__((ext_vector_type(8))) float v8f;
typedef __attribute__((ext_vector_type(4))) unsigned int u32x4;
typedef __attribute__((ext_vector_type(8))) int i32x8;
typedef __attribute__((ext_vector_type(4))) int i32x4;

// ---------------------------- degree kernels -------------------------------

__global__ void k_deg_init(float* __restrict__ deg, int n) {
    int i = blockIdx.x * blockDim.x + threadIdx.x;
    if (i < n) deg[i] = 1.0f;            // self-loop contributes 1
}

__global__ void k_deg_count(const int* __restrict__ col, float* __restrict__ deg, int e) {
    int i = blockIdx.x * blockDim.x + threadIdx.x;
    int stride = gridDim.x * blockDim.x;
    for (; i < e; i += stride) atomicAdd(&deg[col[i]], 1.0f);
}

__global__ void k_deg_rsqrt(float* __restrict__ deg, int n) {
    int i = blockIdx.x * blockDim.x + threadIdx.x;
    if (i < n) deg[i] = rsqrtf(deg[i]);  // deg >= 1 always (self loops)
}

// ----------------------- TDM stripe load (A -> LDS) ------------------------
// DMA a 16-row x 256-col f32 stripe (row stride 256 elems) into LDS with
// 4 dwords of padding after every 256 dwords => LDS row stride 260 floats.
// D# built per CDNA5 ISA 8.3/8.4; tracked with TENSORcnt.
// This toolchain's builtin takes 6 args: (g0, g1, g2, g3, g4, cpol).

__device__ __forceinline__ void tdm_load_stripe(const float* gsrc, unsigned int ldsOff) {
    const unsigned long long ga = (unsigned long long)(uintptr_t)gsrc;
    u32x4 g0;
    g0.x = 1u;                                               // count=1, user D#
    g0.y = ldsOff;                                           // lds_addr (bytes)
    g0.z = (unsigned int)ga;                                 // global_addr[31:0]
    g0.w = ((unsigned int)(ga >> 32) & 0x01FFFFFFu)          // global_addr[56:32]
         | (2u << 30);                                       // type=2 ("image")
    // group1: data_size=4B (2<<16), pad_enable (1<<20),
    //         pad_interval=256 dwords (7<<22), pad_amount=4 dwords (3<<25)
    i32x8 g1 = {
        (2 << 16) | (1 << 20) | (7 << 22) | (3 << 25),       // flags
        (256 << 16),                                         // tensor_dim0 = 256
        (16 << 16),                                          // tensor_dim1 = 16
        (256 << 16),                                         // tile_dim0   = 256
        16,                                                  // tile_dim1   = 16
        256,                                                 // tensor_dim0_stride
        0, 0
    };
    const i32x4 z4 = {0, 0, 0, 0};
    const i32x8 z8 = {0, 0, 0, 0, 0, 0, 0, 0};
    __builtin_amdgcn_tensor_load_to_lds(g0, g1, z4, z4, z8, 0);
}

// ------------------------------ WMMA GEMM ----------------------------------
// Y[M x 256] = X[M x 256] * W[256 x 256].
// Block = 256 threads (8 waves) owns one 16-row stripe; A staged once in LDS
// by the TDM; each wave computes 2 of the 16 column tiles sharing its A frag.
// A 16x4 layout : lanes 0-15 hold K=0,1 (vgpr0,1); lanes 16-31 hold K=2,3.
// B 4x16 layout : lanes 0-15 hold rows K=0,1; lanes 16-31 hold rows K=2,3.
// C 16x16 layout: vgpr j = row (j + 8*half), col = lane & 15.

__global__ void k_gemm_wmma(const float* __restrict__ X,
                            const float* __restrict__ W,
                            float* __restrict__ Y, int M) {
    __shared__ float Atile[16 * LDS_STRIDE];

    const int lane = threadIdx.x & 31;
    const int wave = threadIdx.x >> 5;
    const int m0   = blockIdx.x * 16;              // M is a multiple of 16

    if (wave == 0) {
        tdm_load_stripe(X + (size_t)m0 * C_DIM, (unsigned int)(uintptr_t)Atile);
        __builtin_amdgcn_s_wait_tensorcnt(0);
    }
    __syncthreads();

    const int half = lane >> 4;                    // 0: K=0,1   1: K=2,3
    const int l    = lane & 15;
    const int n0   = wave * 16;                    // column tiles: wave, wave+8
    const int n1   = n0 + 128;

    const float* __restrict__ arow = &Atile[l * LDS_STRIDE];
    v8f acc0 = {}, acc1 = {};

    #pragma unroll 4
    for (int k0 = 0; k0 < C_DIM; k0 += 4) {
        const int ka = k0 + 2 * half;
        v2f a, b0, b1;
        a.x  = arow[ka + 0];
        a.y  = arow[ka + 1];
        b0.x = W[(size_t)(ka + 0) * C_DIM + n0 + l];
        b0.y = W[(size_t)(ka + 1) * C_DIM + n0 + l];
        b1.x = W[(size_t)(ka + 0) * C_DIM + n1 + l];
        b1.y = W[(size_t)(ka + 1) * C_DIM + n1 + l];
        acc0 = __builtin_amdgcn_wmma_f32_16x16x4_f32(
            false, a, false, b0, (short)0, acc0, false, false);
        acc1 = __builtin_amdgcn_wmma_f32_16x16x4_f32(
            false, a, false, b1, (short)0, acc1, false, false);
    }

    float* __restrict__ y0 = Y + (size_t)(m0 + half * 8) * C_DIM + n0 + l;
    #pragma unroll
    for (int j = 0; j < 8; ++j) {
        y0[(size_t)j * C_DIM]       = acc0[j];
        y0[(size_t)j * C_DIM + 128] = acc1[j];
    }
}

// --------------------------- edge scatter-add ------------------------------
// work item = (edge, 4-channel chunk); 64 chunks per edge.

__global__ void k_edge_scatter(const int* __restrict__ row,
                               const int* __restrict__ col,
                               const float* __restrict__ isd,
                               const float* __restrict__ xw,
                               float* __restrict__ agg,
                               int E, int N) {
    const long long total = (long long)(E + N) * 64;
    long long idx = (long long)blockIdx.x * blockDim.x + threadIdx.x;
    const long long stride = (long long)gridDim.x * blockDim.x;
    for (; idx < total; idx += stride) {
        const int e  = (int)(idx >> 6);
        const int ch = ((int)idx & 63) << 2;
        int r, c;
        if (e < E) { r = row[e]; c = col[e]; }
        else       { r = e - E; c = r; }           // self loop
        const float nrm = isd[r] * isd[c];
        const float4 v = *(const float4*)(xw + (size_t)r * C_DIM + ch);
        float* dst = agg + (size_t)c * C_DIM + ch;
        atomicAdd(dst + 0, v.x * nrm);
        atomicAdd(dst + 1, v.y * nrm);
        atomicAdd(dst + 2, v.z * nrm);
        atomicAdd(dst + 3, v.w * nrm);
    }
}

// -------------------------- layernorm + relu -------------------------------
// One wave32 per node; 8 channels per lane (lane + 32*j => coalesced).
// Safe in place (out may alias agg): full row read to registers before write.

__global__ void k_ln_relu(const float* __restrict__ agg,
                          const float* __restrict__ bias,
                          const float* __restrict__ gamma,
                          const float* __restrict__ beta,
                          float* __restrict__ out, int N) {
    const int lane = threadIdx.x & 31;
    const int wave = threadIdx.x >> 5;
    const int node = blockIdx.x * (blockDim.x >> 5) + wave;
    if (node >= N) return;

    const float* a = agg + (size_t)node * C_DIM;
    float h[8];
    float s = 0.0f;
    #pragma unroll
    for (int j = 0; j < 8; ++j) {
        const int ch = lane + 32 * j;
        h[j] = a[ch] + bias[ch];
        s += h[j];
    }
    #pragma unroll
    for (int off = 16; off > 0; off >>= 1) s += __shfl_xor(s, off, 32);
    const float mu = s * (1.0f / C_DIM);

    float vs = 0.0f;
    #pragma unroll
    for (int j = 0; j < 8; ++j) { const float d = h[j] - mu; vs += d * d; }
    #pragma unroll
    for (int off = 16; off > 0; off >>= 1) vs += __shfl_xor(vs, off, 32);
    const float inv = rsqrtf(vs * (1.0f / C_DIM) + 1e-5f);

    float* o = out + (size_t)node * C_DIM;
    #pragma unroll
    for (int j = 0; j < 8; ++j) {
        const int ch = lane + 32 * j;
        const float y = (h[j] - mu) * inv * gamma[ch] + beta[ch];
        o[ch] = fmaxf(y, 0.0f);
    }
}

// ------------------------------- launcher ----------------------------------

extern "C" void kernel_launch(void* const* d_in, const int* in_sizes, int n_in,
                              void* d_out, int out_size, void* d_ws, size_t ws_size,
                              hipStream_t stream) {
    const float* x     = (const float*)d_in[0];
    const int*   edges = (const int*)d_in[1];          // [2, E] row-major
    const float* W1    = (const float*)d_in[2];
    const float* b1    = (const float*)d_in[3];
    const float* W2    = (const float*)d_in[4];
    const float* b2    = (const float*)d_in[5];
    const float* gamma = (const float*)d_in[6];
    const float* beta  = (const float*)d_in[7];

    const int N = in_sizes[0] / C_DIM;                 // 100000
    const int E = in_sizes[1] / 2;                     // 1600000
    const int* row = edges;
    const int* col = edges + E;

    float* z1 = (float*)d_out;                         // [N, C]
    float* z2 = z1 + (size_t)N * C_DIM;                // [N, C]

    // workspace: inv_sqrt_deg (padded) + one xw buffer (~103 MB)
    float* isd = (float*)d_ws;
    float* xw  = isd + (((size_t)N + 255) & ~(size_t)255);

    const size_t featBytes = (size_t)N * C_DIM * sizeof(float);

    // ---- degree normalization -------------------------------------------
    {
        int b256 = 256;
        k_deg_init <<<(N + b256 - 1) / b256, b256, 0, stream>>>(isd, N);
        k_deg_count<<<4096, b256, 0, stream>>>(col, isd, E);
        k_deg_rsqrt<<<(N + b256 - 1) / b256, b256, 0, stream>>>(isd, N);
    }

    const int gemmBlocks = N / 16;                     // one 16-row stripe/block
    const int lnBlocks   = (N + 7) / 8;                // 8 nodes/block

    // ---- layer 1 ---------------------------------------------------------
    k_gemm_wmma<<<gemmBlocks, 256, 0, stream>>>(x, W1, xw, N);
    (void)hipMemsetAsync(z1, 0, featBytes, stream);
    k_edge_scatter<<<8192, 256, 0, stream>>>(row, col, isd, xw, z1, E, N);
    k_ln_relu<<<lnBlocks, 256, 0, stream>>>(z1, b1, gamma, beta, z1, N);

    // ---- layer 2 ---------------------------------------------------------
    k_gemm_wmma<<<gemmBlocks, 256, 0, stream>>>(z1, W2, xw, N);
    (void)hipMemsetAsync(z2, 0, featBytes, stream);
    k_edge_scatter<<<8192, 256, 0, stream>>>(row, col, isd, xw, z2, E, N);
    k_ln_relu<<<lnBlocks, 256, 0, stream>>>(z2, b2, gamma, beta, z2, N);
}